// TimeEmbedding_49246095016509
// MI455X (gfx1250) — compile-verified
//
#include <hip/hip_runtime.h>

// TimeEmbedding: masked forward-fill scan over T, then cos/sin * exp decay
// embedding. T=8192, B=256, D=128 (64 freq pairs). Output [T,B,D] f32.
//
// MI455X (gfx1250, wave32), decoupled two-phase scan for occupancy:
//  Phase 1: per (column, 1024-row segment), last true-event time -> d_ws
//           (256 x 8 floats = 8KB). Position-keyed max reduction.
//  Phase 2: 2048 blocks = (segment, column); each block seeds its carry from
//           earlier segments' summaries, then per 256-row chunk:
//            - intra-wave forward-fill scan via __shfl_up (no barriers)
//            - cross-wave combine via double-buffered 8-entry LDS array
//              -> exactly ONE __syncthreads() per chunk
//            - wave w owns rows [32w,32w+32): d broadcast via v_readlane
//            - lane L owns freq pair (2L,2L+1): exp computed once, reused by
//              cos & sin; two float2 stores/lane -> coalesced 256B bursts
//  CDNA5: weight tables staged with global_load_async_to_lds_b32 +
//         s_wait_asynccnt. Native v_exp/v_sin/v_cos keep the kernel
//         store-bound (~1.07 GB @ 23.3 TB/s ~= 46 us floor).

#define T_DIM   8192
#define B_DIM   256
#define DHALF   64
#define CHUNK   256
#define NWAVE   8
#define NSEG    8
#define SEGLEN  (T_DIM / NSEG)       // 1024

// ---------------- Phase 1: per-segment last-valid summaries ----------------
__global__ __launch_bounds__(256) void seg_summary_kernel(
    const float* __restrict__ tim,
    const unsigned char* __restrict__ mask,
    float* __restrict__ seg_last)            // [B][NSEG], -1 == none
{
    const int seg  = blockIdx.x;
    const int b    = blockIdx.y;
    const int tid  = threadIdx.x;
    const int lane = tid & 31;
    const int wave = tid >> 5;

    __shared__ int   s_pos[NWAVE];
    __shared__ float s_val[NWAVE];

    int   bestp = -1;
    float bestv = -1.0f;
    #pragma unroll
    for (int k = 0; k < SEGLEN / 256; ++k) {
        const int p   = seg * SEGLEN + k * 256 + tid;     // ascending in k
        const int idx = p * B_DIM + b;
        const float tv = tim[idx];
        if (mask[idx]) { bestp = p; bestv = tv; }
    }
    // wave-level max-by-position reduction
    #pragma unroll
    for (int off = 16; off >= 1; off >>= 1) {
        const int   op = __shfl_down(bestp, off, 32);
        const float ov = __shfl_down(bestv, off, 32);
        if (op > bestp) { bestp = op; bestv = ov; }
    }
    if (lane == 0) { s_pos[wave] = bestp; s_val[wave] = bestv; }
    __syncthreads();
    if (tid == 0) {
        int bp = -1; float bv = -1.0f;
        #pragma unroll
        for (int j = 0; j < NWAVE; ++j) {
            if (s_pos[j] > bp) { bp = s_pos[j]; bv = s_val[j]; }
        }
        seg_last[b * NSEG + seg] = (bp >= 0) ? bv : -1.0f;
    }
}

// ---------------- Phase 2: scan + embedding per (segment, column) ----------
__global__ __launch_bounds__(256) void time_embed_kernel(
    const float* __restrict__ tim,
    const unsigned char* __restrict__ mask,   // numpy bool: 1 byte / element
    const float* __restrict__ freq_w,         // [64]
    const float* __restrict__ decay_w,        // [64]
    const float* __restrict__ seg_last,       // [B][NSEG]
    float* __restrict__ out)                  // [T, B, 128]
{
    __shared__ float s_w[DHALF];
    __shared__ float s_a[DHALF];
    __shared__ float s_ws[2][NWAVE];          // double-buffered wave sums

    const int seg  = blockIdx.x;
    const int b    = blockIdx.y;
    const int tid  = threadIdx.x;
    const int lane = tid & 31;
    const int wave = tid >> 5;

    // ---- CDNA5 async global->LDS staging of the two 64-entry weight tables.
    // Low 32 bits of a generic pointer to __shared__ == LDS byte address.
    // The 24-bit instruction offset applies to BOTH memory and LDS addresses,
    // so offset:128 moves elements [32..63] with the same address registers.
    if (tid < 32) {
        unsigned lds_w = (unsigned)(unsigned long long)&s_w[tid];
        unsigned lds_a = (unsigned)(unsigned long long)&s_a[tid];
        unsigned goff  = (unsigned)(tid * 4);
        unsigned long long fp = (unsigned long long)freq_w;
        unsigned long long dp = (unsigned long long)decay_w;
        asm volatile("global_load_async_to_lds_b32 %0, %1, %2 offset:0"
                     :: "v"(lds_w), "v"(goff), "s"(fp) : "memory");
        asm volatile("global_load_async_to_lds_b32 %0, %1, %2 offset:128"
                     :: "v"(lds_w), "v"(goff), "s"(fp) : "memory");
        asm volatile("global_load_async_to_lds_b32 %0, %1, %2 offset:0"
                     :: "v"(lds_a), "v"(goff), "s"(dp) : "memory");
        asm volatile("global_load_async_to_lds_b32 %0, %1, %2 offset:128"
                     :: "v"(lds_a), "v"(goff), "s"(dp) : "memory");
    }
    asm volatile("s_wait_asynccnt 0x0" ::: "memory");
    __syncthreads();

    // Per-lane frequency pair, hoisted out of all loops.
    const int   f0 = lane * 2;
    const float w0 = s_w[f0],        w1 = s_w[f0 + 1];
    const float a0 = fabsf(s_a[f0]), a1 = fabsf(s_a[f0 + 1]);

    // Seed carry from earlier segments' summaries (uniform scalar loads).
    float carry = 0.0f;
    for (int j = 0; j < seg; ++j) {
        const float sl = seg_last[b * NSEG + j];
        carry = (sl >= 0.0f) ? sl : carry;
    }

    for (int c = 0; c < SEGLEN / CHUNK; ++c) {
        // ---- load this chunk's (time, mask) for column b
        const int   row0 = seg * SEGLEN + c * CHUNK;
        const int   idx  = (row0 + tid) * B_DIM + b;
        const float tv   = tim[idx];
        float v = mask[idx] ? tv : -1.0f;         // -1 == "no event here"

        // ---- intra-wave inclusive forward-fill scan (shuffles, no barriers)
        #pragma unroll
        for (int off = 1; off < 32; off <<= 1) {
            const float p = __shfl_up(v, off, 32);
            if (lane >= off && v < 0.0f) v = p;
        }

        // ---- publish wave totals (double buffer), one barrier per chunk
        const int buf = c & 1;
        if (lane == 31) s_ws[buf][wave] = v;
        __syncthreads();

        // exclusive prefix over earlier waves, seeded with the chunk carry
        float pref = carry;
        for (int j = 0; j < wave; ++j) {
            const float wsv = s_ws[buf][j];
            pref = (wsv >= 0.0f) ? wsv : pref;
        }
        // next chunk's carry: fold all 8 wave sums (uniform result)
        float nc = carry;
        #pragma unroll
        for (int j = 0; j < NWAVE; ++j) {
            const float wsv = s_ws[buf][j];
            nc = (wsv >= 0.0f) ? wsv : nc;
        }
        carry = nc;

        // exclusive-within-wave value -> reference time strictly before row
        float excl = __shfl_up(v, 1, 32);
        if (lane == 0) excl = -1.0f;
        const float refp = (excl >= 0.0f) ? excl : pref;
        const float d    = tv - refp;

        // ---- embedding: wave w owns rows [32w, 32w+32); its lanes hold
        //      exactly those d values -> broadcast via v_readlane (SALU path).
        const int dbits = __float_as_int(d);
        float* rowbase = out + (((size_t)(row0 + wave * 32)) * B_DIM + b)
                             * (size_t)(2 * DHALF);
        #pragma unroll 8
        for (int i = 0; i < 32; ++i) {
            const float dv = __int_as_float(__builtin_amdgcn_readlane(dbits, i));
            const float e0 = __expf(-dv * a0);
            const float e1 = __expf(-dv * a1);
            const float p0 = dv * w0, p1 = dv * w1;
            float2 cv, sv;
            cv.x = __cosf(p0) * e0;  cv.y = __cosf(p1) * e1;
            sv.x = __sinf(p0) * e0;  sv.y = __sinf(p1) * e1;
            float* rp = rowbase + (size_t)i * (B_DIM * 2 * DHALF);
            *reinterpret_cast<float2*>(rp + f0)         = cv;   // cos half
            *reinterpret_cast<float2*>(rp + DHALF + f0) = sv;   // sin half
        }
        // no trailing barrier: next chunk uses the other s_ws buffer, and its
        // barrier proves all reads of this buffer finished.
    }
}

extern "C" void kernel_launch(void* const* d_in, const int* in_sizes, int n_in,
                              void* d_out, int out_size, void* d_ws, size_t ws_size,
                              hipStream_t stream) {
    (void)in_sizes; (void)n_in; (void)out_size; (void)ws_size;
    const float*         tim   = (const float*)d_in[0];
    const unsigned char* mask  = (const unsigned char*)d_in[1];   // jnp.bool_
    const float*         freqw = (const float*)d_in[2];
    const float*         decw  = (const float*)d_in[3];
    float*               out   = (float*)d_out;
    float*               segl  = (float*)d_ws;                    // 8 KB

    seg_summary_kernel<<<dim3(NSEG, B_DIM), dim3(256), 0, stream>>>(
        tim, mask, segl);
    time_embed_kernel<<<dim3(NSEG, B_DIM), dim3(256), 0, stream>>>(
        tim, mask, freqw, decw, segl, out);
}